// QuantizedLogSoftmax_12970801234623
// MI455X (gfx1250) — compile-verified
//
#include <hip/hip_runtime.h>
#include <cstdint>

// Problem geometry (fixed by the reference): x is (4, 64, 384, 384) f32.
#define HW        (384 * 384)       // 147456 spatial locations
#define T_STEPS   256               // flattened B*C scan length
#define BLOCK     256               // 8 wave32s per block
#define STAGES    8                 // LDS ring stages (8 KB per block)
#define INFLIGHT  6                 // async loads in flight per wave
#define UNROLL    2                 // scan steps per steady-state iteration

#define QMINF     (-8.0f)
#define QMAXF     (7.9375f)         // 127/16

typedef __attribute__((ext_vector_type(4))) float f32x4;   // native vector for NT builtins

// ---------------------------------------------------------------------------
// CDNA5 async global->LDS copy (ASYNCcnt-tracked) + explicit LDS (AS3) reads.
// ---------------------------------------------------------------------------
typedef __attribute__((address_space(1))) int   gbl_i32;
typedef __attribute__((address_space(3))) int   lds_i32;
typedef __attribute__((address_space(3))) float lds_f32;

__device__ __forceinline__ void async_ld_b32(const float* gptr, float* lds_generic) {
#if __has_builtin(__builtin_amdgcn_global_load_async_to_lds_b32)
    __builtin_amdgcn_global_load_async_to_lds_b32(
        (gbl_i32*)gptr, (lds_i32*)lds_generic, /*offset=*/0, /*cpol=*/0);
#else
    unsigned loff = (unsigned)(unsigned long long)(lds_f32*)lds_generic;
    asm volatile("global_load_async_to_lds_b32 %0, %1, off"
                 :: "v"(loff), "v"(gptr) : "memory");
#endif
}

#if __has_builtin(__builtin_amdgcn_s_wait_asynccnt)
#define WAIT_ASYNCCNT(n) __builtin_amdgcn_s_wait_asynccnt(n)
#else
#define WAIT_ASYNCCNT(n) asm volatile("s_wait_asynccnt %0" :: "i"(n) : "memory")
#endif

// ---------------------------------------------------------------------------
// One scan step: s = clip(max(s, v) + LUT(|s - v|), QMIN, QMAX)
// LUT(d) = round(log1p(exp(-min(floor(d*4096),32767)/4096)) * 65536) / 65536
// Bounded args (d_q in [0,8), e in (3.3e-4,1], 1+e in [1.0003,2]) let us use
// raw v_exp_f32 / v_log_f32 with folded constants:
//   exp(-d_q)   = exp2(d_int * (-log2e/4096))
//   log1p(e)*65536 = log2(1+e) * (ln2 * 65536)
// rintf == round-half-even == jnp.round.
// ---------------------------------------------------------------------------
__device__ __forceinline__ float scan_step(float s, float v) {
    float diff  = fabsf(s - v);
    float d_int = fminf(floorf(diff * 4096.0f), 32767.0f);
    float e     = __builtin_amdgcn_exp2f(d_int * (-1.4426950408889634f / 4096.0f));
    float lu    = rintf(__builtin_amdgcn_logf(1.0f + e) *
                        (0.6931471805599453f * 65536.0f)) * (1.0f / 65536.0f);
    float r     = fmaxf(s, v) + lu;
    return fminf(fmaxf(r, QMINF), QMAXF);
}

// ---------------------------------------------------------------------------
// Pass A: sequential scan over t = 0..255 for each (h,w) column.
// Each lane owns one column and one 4-byte slot per LDS stage, so the async
// pipeline needs no barriers: per-wave s_wait_asynccnt ordering suffices
// (async loads complete in order). Steady state consumes UNROLL stages per
// wait: ASYNCcnt <= INFLIGHT-UNROLL implies the UNROLL oldest loads are done.
// ---------------------------------------------------------------------------
__global__ __launch_bounds__(BLOCK) void
qls_scan_kernel(const float* __restrict__ x, float* __restrict__ sum_exp) {
    __shared__ float stage[STAGES * BLOCK];

    const int  tid = (int)threadIdx.x;
    const long long col = (long long)blockIdx.x * BLOCK + tid;   // h*W + w
    const float* gp = x + col;                                   // stride HW per t

    // AS3 view of this block's staging ring -> ds_load_b32, not flat_load.
    volatile lds_f32* lstage = (volatile lds_f32*)(lds_f32*)stage;

    // Prologue: fill the pipeline.
#pragma unroll
    for (int i = 0; i < INFLIGHT; ++i)
        async_ld_b32(gp + (long long)i * HW, &stage[i * BLOCK + tid]);

    float s = QMINF;

    // Steady state: (T_STEPS - INFLIGHT) = 250 steps, 2 per iteration.
    for (int t = 0; t < T_STEPS - INFLIGHT; t += UNROLL) {
        WAIT_ASYNCCNT(INFLIGHT - UNROLL);
        float v0 = lstage[((t + 0) & (STAGES - 1)) * BLOCK + tid];
        float v1 = lstage[((t + 1) & (STAGES - 1)) * BLOCK + tid];
        async_ld_b32(gp + (long long)(t + INFLIGHT + 0) * HW,
                     &stage[((t + INFLIGHT + 0) & (STAGES - 1)) * BLOCK + tid]);
        async_ld_b32(gp + (long long)(t + INFLIGHT + 1) * HW,
                     &stage[((t + INFLIGHT + 1) & (STAGES - 1)) * BLOCK + tid]);
        s = scan_step(s, v0);
        s = scan_step(s, v1);
    }

    // Drain the last INFLIGHT stages.
    WAIT_ASYNCCNT(0);
#pragma unroll
    for (int t = T_STEPS - INFLIGHT; t < T_STEPS; ++t) {
        float v = lstage[(t & (STAGES - 1)) * BLOCK + tid];
        s = scan_step(s, v);
    }

    sum_exp[col] = s;
}

// ---------------------------------------------------------------------------
// Pass B: out = clip(round((x - sum_exp[hw]) * 16), -128, 127) / 16.
// Grid = (HW/(BLOCK*4)) x T_STEPS so hw is affine in IDs (no 64-bit modulo).
// x is the last use (NT load); out is write-once (NT store) so the 151 MB
// output stream doesn't thrash the 192 MB L2.
// ---------------------------------------------------------------------------
__device__ __forceinline__ float qfeat(float d) {
    return fminf(fmaxf(rintf(d * 16.0f), -128.0f), 127.0f) * 0.0625f;
}

__global__ __launch_bounds__(BLOCK) void
qls_quant_kernel(const float* __restrict__ x, const float* __restrict__ sum_exp,
                 float* __restrict__ out) {
    const int hw = (blockIdx.x * BLOCK + threadIdx.x) * 4;           // < HW
    const long long i = (long long)blockIdx.y * HW + hw;             // element index

    const f32x4 xv = __builtin_nontemporal_load((const f32x4*)(x + i));
    const f32x4 sv = *(const f32x4*)(sum_exp + hw);   // heavy reuse: keep cached

    f32x4 o;
    o.x = qfeat(xv.x - sv.x);
    o.y = qfeat(xv.y - sv.y);
    o.z = qfeat(xv.z - sv.z);
    o.w = qfeat(xv.w - sv.w);
    __builtin_nontemporal_store(o, (f32x4*)(out + i));
}

// ---------------------------------------------------------------------------
extern "C" void kernel_launch(void* const* d_in, const int* in_sizes, int n_in,
                              void* d_out, int out_size, void* d_ws, size_t ws_size,
                              hipStream_t stream) {
    const float* x   = (const float*)d_in[0];
    float*       out = (float*)d_out;
    float*       se  = (float*)d_ws;     // HW floats = 576 KB scratch

    // Pass A: 147456 columns / 256 threads = 576 blocks.
    qls_scan_kernel<<<HW / BLOCK, BLOCK, 0, stream>>>(x, se);

    // Pass B: x = 144 blocks over (h,w)/4, y = 256 over flattened (b,c).
    dim3 gridB(HW / (BLOCK * 4), T_STEPS, 1);
    qls_quant_kernel<<<gridB, BLOCK, 0, stream>>>(x, se, out);
}